// Ema_39685497815535
// MI455X (gfx1250) — compile-verified
//
#include <hip/hip_runtime.h>
#include <hip/hip_bf16.h>

typedef float v2f __attribute__((ext_vector_type(2)));
typedef float v8f __attribute__((ext_vector_type(8)));

typedef __attribute__((address_space(1))) int gint;   // global (addrspace 1)
typedef __attribute__((address_space(3))) int lint;   // LDS    (addrspace 3)

#define CTILE   32          // channels per block (== waves per block)
#define TTILE   256         // timesteps per macro-tile = 16 blocks x 16 steps
#define CPAD    33          // LDS row pitch (floats), odd to spread banks
#define THREADS 1024        // 32 waves of 32 lanes
#define LOADS_PER_THREAD ((TTILE * CTILE) / THREADS)   // 8

#if __has_builtin(__builtin_amdgcn_global_load_async_to_lds_b32) && \
    __has_builtin(__builtin_amdgcn_s_wait_asynccnt)
#define EMA_ASYNC 1
#else
#define EMA_ASYNC 0
#endif

__device__ __forceinline__ float lane_bcast(float v, int srcLane) {
#if __has_builtin(__builtin_amdgcn_readlane)
  return __int_as_float(__builtin_amdgcn_readlane(__float_as_int(v), srcLane));
#else
  return __shfl(v, srcLane, 32);
#endif
}

__global__ __launch_bounds__(THREADS)
void ema_scan_kernel(const float* __restrict__ xg,
                     const float* __restrict__ drg,
                     float* __restrict__ yg,
                     int B, int T, int C) {
  __shared__ float xs[2][TTILE * CPAD];   // double-buffered input tile
  __shared__ float ys[TTILE * CPAD];      // output staging tile

  const int tid  = threadIdx.x;
  const int lane = tid & 31;
  const int wave = tid >> 5;              // wave == local channel index

  const int nct = C / CTILE;
  const int b   = blockIdx.x / nct;
  const int c0  = (blockIdx.x % nct) * CTILE;

  // ---- per-wave decay factor: f = exp(-softplus(dr) * 16 / T) ----
  const int   c    = c0 + wave;
  const float dr   = drg[c];
  const float sp   = fmaxf(dr, 0.0f) + log1pf(__expf(-fabsf(dr)));  // stable softplus
  const float rate = sp * (16.0f / (float)T);                       // f^p = exp(-rate*p)

  // ---- A matrix = lower-triangular L[i][j] = f^(i-j), as 4 K=4 slices ----
  // A 16x4 f32 VGPR layout: v0 = {K=0 | K=2}, v1 = {K=1 | K=3} over lane halves.
  const int M  = lane & 15;
  const int hi = lane >> 4;               // 0: lanes 0-15, 1: lanes 16-31
  v2f a[4];
  #pragma unroll
  for (int k = 0; k < 4; ++k) {
    const int ka0 = 4 * k + 2 * hi;
    const int ka1 = ka0 + 1;
    a[k].x = (M >= ka0) ? __expf(-rate * (float)(M - ka0)) : 0.0f;
    a[k].y = (M >= ka1) ? __expf(-rate * (float)(M - ka1)) : 0.0f;
  }
  const float f16p = __expf(-rate * 16.0f);   // f^16, per-block-column carry factor
  float pwm[8];                                // f^(Mabs+1) for the carry fix-up
  #pragma unroll
  for (int m = 0; m < 8; ++m)
    pwm[m] = __expf(-rate * (float)(m + 8 * hi + 1));

  const size_t strip  = ((size_t)b * T) * C + c0;
  const int    ntiles = T / TTILE;

  // ---- coalesced (async) global -> LDS tile copy ----
  auto issue_tile = [&](int tile, int buf) {
    const float* src_base = xg + strip + (size_t)tile * TTILE * C;
    #pragma unroll
    for (int i = 0; i < LOADS_PER_THREAD; ++i) {
      const int idx  = tid + i * THREADS;
      const int t_in = idx >> 5;           // CTILE == 32
      const int cl   = idx & 31;
      const float* src = src_base + (size_t)t_in * C + cl;
      float*       dst = &xs[buf][t_in * CPAD + cl];
#if EMA_ASYNC
      __builtin_amdgcn_global_load_async_to_lds_b32((gint*)src, (lint*)dst, 0, 0);
#else
      *dst = *src;
#endif
    }
  };

  issue_tile(0, 0);
  float y_carry = 0.0f;                    // scan state entering current macro-tile

  for (int tile = 0; tile < ntiles; ++tile) {
    const int cur = tile & 1;
    if (tile + 1 < ntiles) {
      issue_tile(tile + 1, cur ^ 1);       // prefetch next tile into other buffer
#if EMA_ASYNC
      __builtin_amdgcn_s_wait_asynccnt(LOADS_PER_THREAD);  // wait only for current tile
#endif
    } else {
#if EMA_ASYNC
      __builtin_amdgcn_s_wait_asynccnt(0);
#endif
    }
    __syncthreads();

    // ---- intra-block scan via f32 WMMA: D = L x X (16 steps x 16 blocks) ----
    // B 4x16 f32 layout mirrors A: v0 = {K=0 | K=2}, v1 = {K=1 | K=3}, N = lane&15.
    const float* xt = xs[cur];
    const int n = lane & 15;               // block column index (D: N = lane&15)
    v8f d = {0.f, 0.f, 0.f, 0.f, 0.f, 0.f, 0.f, 0.f};
    #pragma unroll
    for (int k = 0; k < 4; ++k) {
      const int kb = 4 * k + 2 * hi;
      v2f bv;
      bv.x = xt[(n * 16 + kb    ) * CPAD + wave];
      bv.y = xt[(n * 16 + kb + 1) * CPAD + wave];
      d = __builtin_amdgcn_wmma_f32_16x16x4_f32(false, a[k], false, bv,
                                                (short)0, d, false, false);
    }

    // ---- cross-column carry scan: cin(n) = f^16*cin(n-1) + D[15][n-1] ----
    // D[15][nn] lives in lane 16+nn of d[7]; uniform source lane -> v_readlane.
    float cin = 0.0f;
    float run = y_carry;
    #pragma unroll
    for (int nn = 0; nn < 16; ++nn) {
      if (n == nn) cin = run;
      const float dlast = lane_bcast(d[7], 16 + nn);
      run = f16p * run + dlast;
    }
    y_carry = run;

    // ---- fix-up + stage result: Y[M][n] = D[M][n] + f^(M+1)*cin(n) ----
    #pragma unroll
    for (int m = 0; m < 8; ++m) {
      const float val  = d[m] + pwm[m] * cin;
      const int   t_in = n * 16 + m + 8 * hi;
      ys[t_in * CPAD + wave] = val;
    }
    __syncthreads();

    // ---- coalesced non-temporal store LDS -> global (y never re-read) ----
    float* dst_base = yg + strip + (size_t)tile * TTILE * C;
    #pragma unroll
    for (int i = 0; i < LOADS_PER_THREAD; ++i) {
      const int idx  = tid + i * THREADS;
      const int t_in = idx >> 5;
      const int cl   = idx & 31;
      __builtin_nontemporal_store(ys[t_in * CPAD + cl],
                                  dst_base + (size_t)t_in * C + cl);
    }
    __syncthreads();                        // free ys and xs[cur] for reuse
  }
}

extern "C" void kernel_launch(void* const* d_in, const int* in_sizes, int n_in,
                              void* d_out, int out_size, void* d_ws, size_t ws_size,
                              hipStream_t stream) {
  const float* x  = (const float*)d_in[0];   // [B, T, C] f32
  const float* dr = (const float*)d_in[1];   // [C] f32
  float*       y  = (float*)d_out;           // [B, T, C] f32

  const int C = in_sizes[1];                 // 1024
  const int T = 4096;
  const int B = in_sizes[0] / (T * C);       // 8

  const int blocks = B * (C / CTILE);        // 256 blocks x 1024 threads
  ema_scan_kernel<<<blocks, THREADS, 0, stream>>>(x, dr, y, B, T, C);
}